// GCN_MSSK_34368328302746
// MI455X (gfx1250) — compile-verified
//
#include <hip/hip_runtime.h>

typedef float v2f __attribute__((ext_vector_type(2)));
typedef float v8f __attribute__((ext_vector_type(8)));

static constexpr int NN = 20000;   // nodes per graph
static constexpr int NE = 320000;  // edges per graph

// ---------------------------------------------------------------- utilities
__global__ void k_zero(float* __restrict__ p, int n) {
  for (int i = blockIdx.x * blockDim.x + threadIdx.x; i < n;
       i += gridDim.x * blockDim.x)
    p[i] = 0.0f;
}

// agg[dst[e]*lda + f] += ew[e] * x[src[e]*ldx + f], f in [0,F). FT = pow2 >= F.
template <int FT, int SHIFT>
__global__ void k_scatter(const int* __restrict__ src, const int* __restrict__ dst,
                          const float* __restrict__ ew, const float* __restrict__ x,
                          int F, int ldx, float* __restrict__ agg, int lda) {
  const int total = NE * FT;
  for (int i = blockIdx.x * blockDim.x + threadIdx.x; i < total;
       i += gridDim.x * blockDim.x) {
    int e = i >> SHIFT;
    int f = i & (FT - 1);
    if (f < F) {
      float v = ew[e] * x[src[e] * ldx + f];
      atomicAdd(&agg[dst[e] * lda + f], v);
    }
  }
}

// Copy W1m (17x128) into a zero-padded 20x128 buffer so K%4==0 for WMMA.
__global__ void k_padW(const float* __restrict__ W, float* __restrict__ Wp) {
  int i = blockIdx.x * blockDim.x + threadIdx.x;
  if (i >= 20 * 128) return;
  int r = i >> 7, c = i & 127;
  Wp[i] = (r < 17) ? W[r * 128 + c] : 0.0f;
}

// ------------------------------------------------------------ WMMA fp32 GEMM
// C[MxN] = relu?(A[MxK] @ B[KxN] + bias), exact fp32 via V_WMMA_F32_16X16X4_F32.
// One wave32 per (column tile x row-tile chunk): all B fragments for the
// column tile are preloaded into registers once (KS x v2f) and reused across
// RPW row tiles. The fully-unrolled inner loop issues one clause-batched
// aligned global_load_b64 (A fragment) per WMMA; bias is folded into the
// accumulator init; RELU is a compile-time flag (no per-element cndmask).
// Layouts per ISA 7.12.2: A frag lane half h, idx=lane&15 -> A[idx][2h+j];
// B mirrored; C/D 8 VGPRs with row = r + 8h, col = idx.
template <int K, int RPW, bool RELU>
__global__ void k_gemm_wmma(const float* __restrict__ A, const float* __restrict__ B,
                            const float* __restrict__ bias, float* __restrict__ C,
                            int M, int N) {
  constexpr int KS = K / 4;
  int lane = threadIdx.x & 31;
  int wid  = blockIdx.x * (blockDim.x >> 5) + (threadIdx.x >> 5);
  int colTiles  = N >> 4;
  int rowChunks = (M >> 4) / RPW;   // M/16 divisible by RPW by construction
  if (wid >= colTiles * rowChunks) return;  // wave-uniform: EXEC all-1s for WMMA
  int ct = wid % colTiles;
  int rc = wid / colTiles;
  int half = lane >> 4;
  int idx  = lane & 15;
  int n = ct * 16 + idx;

  // Preload B fragments for this column tile (once per wave).
  v2f bfrag[KS];
#pragma unroll
  for (int kk = 0; kk < KS; ++kk) {
    int ka = kk * 4 + half * 2;
    v2f b;
    b[0] = B[ka * N + n];
    b[1] = B[(ka + 1) * N + n];
    bfrag[kk] = b;
  }
  float bb = bias[n];

  for (int r0 = rc * RPW; r0 < rc * RPW + RPW; ++r0) {
    const float* __restrict__ arow = A + (r0 * 16 + idx) * K + half * 2;
    // Fold bias into the accumulator: C = A@B + b (broadcast over rows).
    v8f acc = {bb, bb, bb, bb, bb, bb, bb, bb};
#pragma unroll
    for (int kk = 0; kk < KS; ++kk) {
      v2f av = *(const v2f*)(arow + kk * 4);  // 8B-aligned for K in {20,128}
      acc = __builtin_amdgcn_wmma_f32_16x16x4_f32(false, av, false, bfrag[kk],
                                                  (short)0, acc, false, false);
    }
    int rowBase = r0 * 16 + half * 8;
#pragma unroll
    for (int r = 0; r < 8; ++r) {
      float v = acc[r];
      if (RELU) v = fmaxf(v, 0.0f);
      C[(rowBase + r) * N + n] = v;
    }
  }
}

// Column mean over NN rows, multi-block: each block sums a row stripe
// (coalesced across the F=256 columns) and atomically combines /NN.
// `out` must be zeroed beforehand.
__global__ void k_colmean(const float* __restrict__ H, int F, float* __restrict__ out) {
  int j = threadIdx.x;
  if (j >= F) return;
  float s = 0.0f;
  for (int r = blockIdx.x; r < NN; r += gridDim.x) s += H[r * F + j];
  atomicAdd(&out[j], s * (1.0f / (float)NN));
}

// features = [hg_seq(256) | hg_mol(256) | smiles(574)]  (first 1086 to match Wfc1)
__global__ void k_feat(const float* __restrict__ hg_seq, const float* __restrict__ hg_mol,
                       const float* __restrict__ smiles, float* __restrict__ feat) {
  int i = blockIdx.x * blockDim.x + threadIdx.x;
  if (i >= 1086) return;
  float v;
  if (i < 256)       v = hg_seq[i];
  else if (i < 512)  v = hg_mol[i - 256];
  else               v = smiles[i - 512];
  feat[i] = v;
}

// Whole MLP head in one block: 1086->575->256->64->1 (tiny: ~0.8 MFLOP).
__global__ void k_mlp(const float* __restrict__ feat,
                      const float* __restrict__ W1, const float* __restrict__ b1,
                      const float* __restrict__ W2, const float* __restrict__ b2,
                      const float* __restrict__ W3, const float* __restrict__ b3,
                      const float* __restrict__ W4, const float* __restrict__ b4,
                      float* __restrict__ out) {
  __shared__ float sf[1086];
  __shared__ float h1[575];
  __shared__ float h2[256];
  __shared__ float h3[64];
  int t = threadIdx.x;
  for (int i = t; i < 1086; i += blockDim.x) sf[i] = feat[i];
  __syncthreads();
  if (t < 575) {
    float s = b1[t];
    for (int i = 0; i < 1086; ++i) s += sf[i] * W1[i * 575 + t];
    h1[t] = fmaxf(s, 0.0f);
  }
  __syncthreads();
  if (t < 256) {
    float s = b2[t];
    for (int i = 0; i < 575; ++i) s += h1[i] * W2[i * 256 + t];
    h2[t] = fmaxf(s, 0.0f);
  }
  __syncthreads();
  if (t < 64) {
    float s = b3[t];
    for (int i = 0; i < 256; ++i) s += h2[i] * W3[i * 64 + t];
    h3[t] = fmaxf(s, 0.0f);
  }
  __syncthreads();
  if (t == 0) {
    float s = b4[0];
    for (int i = 0; i < 64; ++i) s += h3[i] * W4[i];
    out[0] = s;
  }
}

// ------------------------------------------------------------------- driver
extern "C" void kernel_launch(void* const* d_in, const int* in_sizes, int n_in,
                              void* d_out, int out_size, void* d_ws, size_t ws_size,
                              hipStream_t stream) {
  const float* feat_mol = (const float*)d_in[0];
  const float* feat_seq = (const float*)d_in[1];
  const float* smiles   = (const float*)d_in[2];
  const float* w_mol    = (const float*)d_in[3];
  const float* w_seq    = (const float*)d_in[4];
  const int*   src_mol  = (const int*)d_in[5];
  const int*   dst_mol  = (const int*)d_in[6];
  const int*   src_seq  = (const int*)d_in[7];
  const int*   dst_seq  = (const int*)d_in[8];
  const float* W1m = (const float*)d_in[9];
  const float* b1m = (const float*)d_in[10];
  const float* W2m = (const float*)d_in[11];
  const float* b2m = (const float*)d_in[12];
  const float* W1s = (const float*)d_in[13];
  const float* b1s = (const float*)d_in[14];
  const float* W2s = (const float*)d_in[15];
  const float* b2s = (const float*)d_in[16];
  const float* Wfc1 = (const float*)d_in[17];
  const float* bfc1 = (const float*)d_in[18];
  const float* Wfc2 = (const float*)d_in[19];
  const float* bfc2 = (const float*)d_in[20];
  const float* Wfc3 = (const float*)d_in[21];
  const float* bfc3 = (const float*)d_in[22];
  const float* Wfc4 = (const float*)d_in[23];
  const float* bfc4 = (const float*)d_in[24];

  float* ws     = (float*)d_ws;
  float* bufA   = ws;                    // scatter target, reused:    20000*128
  float* bufH1  = bufA  + NN * 128;      // layer-1 activations:       20000*128
  float* bufH2  = bufH1 + NN * 128;      // layer-2 activations:       20000*256
  float* hg_mol = bufH2 + NN * 256;      // 256
  float* hg_seq = hg_mol + 256;          // 256
  float* W1mp   = hg_seq + 256;          // 20*128 zero-padded W1m
  float* featv  = W1mp + 20 * 128;       // 1086

  dim3 b256(256);

  k_padW<<<dim3((20 * 128 + 255) / 256), b256, 0, stream>>>(W1m, W1mp);
  k_zero<<<dim3(1), b256, 0, stream>>>(hg_mol, 512);  // hg_mol + hg_seq

  // ---- molecule graph: scatter(17-pad-20) -> WMMA -> scatter(128) -> WMMA -> mean
  k_zero<<<dim3(1024), b256, 0, stream>>>(bufA, NN * 20);
  k_scatter<32, 5><<<dim3(8192), b256, 0, stream>>>(src_mol, dst_mol, w_mol,
                                                    feat_mol, 17, 17, bufA, 20);
  // rowChunks = 1250/10 = 125; waves = 125*8 = 1000; blocks = 125
  k_gemm_wmma<20, 10, true><<<dim3(125), b256, 0, stream>>>(bufA, W1mp, b1m,
                                                            bufH1, NN, 128);
  k_zero<<<dim3(1024), b256, 0, stream>>>(bufA, NN * 128);
  k_scatter<128, 7><<<dim3(16384), b256, 0, stream>>>(src_mol, dst_mol, w_mol,
                                                      bufH1, 128, 128, bufA, 128);
  // waves = 125*16 = 2000; blocks = 250
  k_gemm_wmma<128, 10, false><<<dim3(250), b256, 0, stream>>>(bufA, W2m, b2m,
                                                              bufH2, NN, 256);
  k_colmean<<<dim3(64), b256, 0, stream>>>(bufH2, 256, hg_mol);

  // ---- sequence graph
  k_zero<<<dim3(1024), b256, 0, stream>>>(bufA, NN * 128);
  k_scatter<128, 7><<<dim3(16384), b256, 0, stream>>>(src_seq, dst_seq, w_seq,
                                                      feat_seq, 128, 128, bufA, 128);
  k_gemm_wmma<128, 10, true><<<dim3(125), b256, 0, stream>>>(bufA, W1s, b1s,
                                                             bufH1, NN, 128);
  k_zero<<<dim3(1024), b256, 0, stream>>>(bufA, NN * 128);
  k_scatter<128, 7><<<dim3(16384), b256, 0, stream>>>(src_seq, dst_seq, w_seq,
                                                      bufH1, 128, 128, bufA, 128);
  k_gemm_wmma<128, 10, false><<<dim3(250), b256, 0, stream>>>(bufA, W2s, b2s,
                                                              bufH2, NN, 256);
  k_colmean<<<dim3(64), b256, 0, stream>>>(bufH2, 256, hg_seq);

  // ---- head
  k_feat<<<dim3((1086 + 255) / 256), b256, 0, stream>>>(hg_seq, hg_mol, smiles, featv);
  k_mlp<<<dim3(1), dim3(640), 0, stream>>>(featv, Wfc1, bfc1, Wfc2, bfc2,
                                           Wfc3, bfc3, Wfc4, bfc4, (float*)d_out);
}